// MultiHeadAttention_57406532879026
// MI455X (gfx1250) — compile-verified
//
#include <hip/hip_runtime.h>
#include <math.h>

// ---------------- problem constants (fixed by the reference) ----------------
#define DMODEL 2048
#define BATCH  16
#define HEADS  16
#define DHEAD  128
#define SEQT   4096   // start_position + S
#define NEWT   4095   // index of the freshly projected k/v row

typedef float v2f __attribute__((ext_vector_type(2)));
typedef float v8f __attribute__((ext_vector_type(8)));

// ---------------------------------------------------------------------------
// fp32 WMMA tile: Y[16 x 2048] = X[16 x 2048] @ W[2048 x 2048]
// One wave per 16x16 output tile, K-loop in steps of 4 via
// V_WMMA_F32_16X16X4_F32 (full fp32 precision; workload is memory-bound so
// the matrix pipe is nowhere near saturated — precision is free here).
//
// Fragment layouts (ISA 7.12.2):
//   A 16x4 : lanes 0-15 -> M=lane, VGPR j = K=j      ; lanes 16-31 -> K=2+j
//   B 4x16 : lanes      -> N=lane&15, VGPR j -> K = (lane>>4)*2 + j
//   C 16x16: VGPR i -> M = i + (lane>>4)*8, N = lane&15
// ---------------------------------------------------------------------------
__device__ __forceinline__ void proj_tile_f32(const float* __restrict__ X,
                                              const float* __restrict__ W,
                                              float* __restrict__ Y,
                                              int n0, int lane) {
  const int col = lane & 15;
  const int kh  = (lane >> 4) * 2;   // 0 or 2
  v8f c = {};
  for (int k = 0; k < DMODEL; k += 4) {
    // A fragment: two consecutive fp32 from row `col`
    v2f a = *(const v2f*)(X + (size_t)col * DMODEL + k + kh);
    // B fragment: two rows of W, same column
    v2f b;
    b.x = W[(size_t)(k + kh + 0) * DMODEL + n0 + col];
    b.y = W[(size_t)(k + kh + 1) * DMODEL + n0 + col];
    c = __builtin_amdgcn_wmma_f32_16x16x4_f32(
            /*neg_a=*/false, a, /*neg_b=*/false, b,
            /*c_mod=*/(short)0, c, /*reuse_a=*/false, /*reuse_b=*/false);
  }
  const int mh = (lane >> 4) * 8;
#pragma unroll
  for (int i = 0; i < 8; ++i)
    Y[(size_t)(mh + i) * DMODEL + n0 + col] = c[i];
}

// Fused Q/K/V projection: 384 blocks = 3 weights x 128 N-tiles, 1 wave each.
__global__ __launch_bounds__(32) void qkv_wmma(const float* __restrict__ X,
                                               const float* __restrict__ Wq,
                                               const float* __restrict__ Wk,
                                               const float* __restrict__ Wv,
                                               float* __restrict__ q,
                                               float* __restrict__ k,
                                               float* __restrict__ v) {
  const int which = blockIdx.x >> 7;          // 0,1,2
  const int n0    = (blockIdx.x & 127) * 16;
  const float* W = (which == 0) ? Wq : (which == 1) ? Wk : Wv;
  float*       Y = (which == 0) ? q  : (which == 1) ? k  : v;
  proj_tile_f32(X, W, Y, n0, (int)threadIdx.x);
}

// Output projection: 128 blocks, 1 wave each.
__global__ __launch_bounds__(32) void proj16_wmma(const float* __restrict__ X,
                                                  const float* __restrict__ W,
                                                  float* __restrict__ Y) {
  proj_tile_f32(X, W, Y, blockIdx.x * 16, (int)threadIdx.x);
}

// ---------------------------------------------------------------------------
// Decode attention for one (b,h) pair per 256-thread block (8 wave32).
// RoPE applied on the fly to q and to the new k row; input caches are never
// written (t == NEWT is served from LDS copies instead).  The special row is
// handled OUTSIDE the streaming loops so the hot paths are pure load+FMA.
// ---------------------------------------------------------------------------
__global__ __launch_bounds__(256) void attn_decode(
    const float* __restrict__ Kc, const float* __restrict__ Vc,
    const float* __restrict__ q_ws, const float* __restrict__ k_ws,
    const float* __restrict__ v_ws, const float* __restrict__ fcos,
    const float* __restrict__ fsin, float* __restrict__ out_ws) {
  __shared__ __align__(16) float sc[SEQT];           // 16 KB scores
  __shared__ __align__(16) float qs[DHEAD];
  __shared__ __align__(16) float kn[DHEAD];
  __shared__ __align__(16) float vn[DHEAD];
  __shared__ float red[256];

  const int b   = blockIdx.x >> 4;
  const int h   = blockIdx.x & 15;
  const int tid = threadIdx.x;
  const float scale = 0.08838834764831845f;          // 1/sqrt(128)

  const float* qp = q_ws + (size_t)b * DMODEL + h * DHEAD;
  const float* kp = k_ws + (size_t)b * DMODEL + h * DHEAD;
  const float* vp = v_ws + (size_t)b * DMODEL + h * DHEAD;

  // ---- RoPE(q), RoPE(k_new), copy v_new into LDS ----
  if (tid < 64) {
    const float c = fcos[tid], s = fsin[tid];
    const float xr = qp[2 * tid], xi = qp[2 * tid + 1];
    qs[2 * tid]     = xr * c - xi * s;
    qs[2 * tid + 1] = xr * s + xi * c;
  } else if (tid < 128) {
    const int p = tid - 64;
    const float c = fcos[p], s = fsin[p];
    const float xr = kp[2 * p], xi = kp[2 * p + 1];
    kn[2 * p]     = xr * c - xi * s;
    kn[2 * p + 1] = xr * s + xi * c;
  } else {
    vn[tid - 128] = vp[tid - 128];
  }
  __syncthreads();

  // ---- Phase A: scores.  Each wave streams TWO adjacent 512 B key rows per
  //      iteration (two clause-able global_load_b128 per lane -> more loads
  //      in flight), then two butterfly shuffle reductions. ----
  const int wave = tid >> 5;
  const int lane = tid & 31;
  const float4 q4 = ((const float4*)qs)[lane];
  const float* kbase = Kc + (((size_t)b * SEQT) * HEADS + h) * DHEAD;

  // pairs (t, t+1), t = 2*wave + 16*i ; covers 0..4094 even starts;
  // the pair (4094,4095) is excluded (t+1 == NEWT) and handled below.
  for (int t = wave * 2; t + 1 < NEWT; t += 16) {
    const float* row0 = kbase + (size_t)t * HEADS * DHEAD;
    const float* row1 = row0 + (size_t)HEADS * DHEAD;
    const float4 k0 = ((const float4*)row0)[lane];
    const float4 k1 = ((const float4*)row1)[lane];
    if (t + 128 < NEWT)   // stay ahead of the HBM stream
      __builtin_prefetch(kbase + (size_t)(t + 128) * HEADS * DHEAD + lane * 4, 0, 0);
    float p0 = k0.x * q4.x + k0.y * q4.y + k0.z * q4.z + k0.w * q4.w;
    float p1 = k1.x * q4.x + k1.y * q4.y + k1.z * q4.z + k1.w * q4.w;
#pragma unroll
    for (int off = 16; off > 0; off >>= 1) {
      p0 += __shfl_xor(p0, off, 32);
      p1 += __shfl_xor(p1, off, 32);
    }
    if (lane == 0) { sc[t] = p0 * scale; sc[t + 1] = p1 * scale; }
  }
  // tail rows: 4094 (global, wave 0) and 4095 == NEWT (LDS kn, wave 1)
  if (wave == 0) {
    const float4 k0 = ((const float4*)(kbase + (size_t)(NEWT - 1) * HEADS * DHEAD))[lane];
    float p0 = k0.x * q4.x + k0.y * q4.y + k0.z * q4.z + k0.w * q4.w;
#pragma unroll
    for (int off = 16; off > 0; off >>= 1) p0 += __shfl_xor(p0, off, 32);
    if (lane == 0) sc[NEWT - 1] = p0 * scale;
  } else if (wave == 1) {
    const float4 k0 = ((const float4*)kn)[lane];
    float p0 = k0.x * q4.x + k0.y * q4.y + k0.z * q4.z + k0.w * q4.w;
#pragma unroll
    for (int off = 16; off > 0; off >>= 1) p0 += __shfl_xor(p0, off, 32);
    if (lane == 0) sc[NEWT] = p0 * scale;
  }
  __syncthreads();

  // ---- softmax: block-reduced max, then exp + sum; normalization folded
  //      into the final output scale. ----
  float m = -3.402823466e38f;
  for (int t = tid; t < SEQT; t += 256) m = fmaxf(m, sc[t]);
  red[tid] = m;
  __syncthreads();
  for (int s = 128; s > 0; s >>= 1) {
    if (tid < s) red[tid] = fmaxf(red[tid], red[tid + s]);
    __syncthreads();
  }
  m = red[0];
  __syncthreads();

  float lsum = 0.f;
  for (int t = tid; t < SEQT; t += 256) {
    const float e = __expf(sc[t] - m);
    sc[t] = e;
    lsum += e;
  }
  red[tid] = lsum;
  __syncthreads();
  for (int s = 128; s > 0; s >>= 1) {
    if (tid < s) red[tid] += red[tid + s];
    __syncthreads();
  }
  const float inv = 1.0f / red[0];
  __syncthreads();   // before reusing red[]

  // ---- Phase B: out[d] = inv * sum_t e[t] * V[t][d].
  //      128 threads own one d each -> every timestep's V access is one
  //      contiguous coalesced 512 B row; two groups split T.  The NEWT row
  //      (LDS) is added once, outside the streaming loop. ----
  const int d = tid & 127;
  const int g = tid >> 7;
  const float* vbase = Vc + (((size_t)b * SEQT) * HEADS + h) * DHEAD;
  float acc = 0.f;
  const int t0 = g * (SEQT / 2);
  const int t1 = (t0 + (SEQT / 2) < NEWT) ? t0 + (SEQT / 2) : NEWT;
#pragma unroll 4
  for (int t = t0; t < t1; ++t)
    acc += sc[t] * vbase[(size_t)t * HEADS * DHEAD + d];
  if (g == 1) acc += sc[NEWT] * vn[d];    // fresh v row from LDS
  red[tid] = acc;
  __syncthreads();
  if (g == 0)
    out_ws[(size_t)b * DMODEL + h * DHEAD + d] = (red[tid] + red[tid + 128]) * inv;
}

// ---------------------------------------------------------------------------
extern "C" void kernel_launch(void* const* d_in, const int* in_sizes, int n_in,
                              void* d_out, int out_size, void* d_ws, size_t ws_size,
                              hipStream_t stream) {
  (void)in_sizes; (void)n_in; (void)out_size; (void)ws_size;
  const float* x  = (const float*)d_in[0];   // inputs      (16,1,2048)
  const float* kc = (const float*)d_in[1];   // key_cache   (16,4096,16,128)
  const float* vc = (const float*)d_in[2];   // value_cache (16,4096,16,128)
  const float* fc = (const float*)d_in[3];   // freqs_cos   (1,64)
  const float* fs = (const float*)d_in[4];   // freqs_sin   (1,64)
  const float* Wq = (const float*)d_in[5];
  const float* Wk = (const float*)d_in[6];
  const float* Wv = (const float*)d_in[7];
  const float* Wo = (const float*)d_in[8];
  float* out = (float*)d_out;                // (16,1,2048) fp32

  float* ws = (float*)d_ws;                  // 4 x 32768 fp32 panels
  float* qw = ws;
  float* kw = ws + 32768;
  float* vw = ws + 65536;
  float* aw = ws + 98304;

  qkv_wmma   <<<384, 32, 0, stream>>>(x, Wq, Wk, Wv, qw, kw, vw);
  attn_decode<<<256, 256, 0, stream>>>(kc, vc, qw, kw, vw, fc, fs, aw);
  proj16_wmma<<<128, 32, 0, stream>>>(aw, Wo, out);
}